// OutputLayer_53824530154128
// MI455X (gfx1250) — compile-verified
//
#include <hip/hip_runtime.h>
#include <hip/hip_bf16.h>

// Native clang vector type: accepted by __builtin_nontemporal_* and lowers
// to 128-bit global_load_b128 / global_store_b128 on gfx1250.
typedef __attribute__((ext_vector_type(4))) float v4f;

// Problem shape (fixed by the reference setup).
#define E_ 16
#define B_ 4096
#define D_ 1024
#define F4_PER_ROW (D_ / 4)            // 256 v4f per row
#define F4_PER_LANE (F4_PER_ROW / 32)  // 8 v4f per lane (wave32)
#define WAVES_PER_BLOCK 8

// One wave32 per output row b:
//  1) every lane redundantly computes argmax over the 16 weights of row b
//     (64B = 1 cacheline; broadcast from cache, no LDS / no sync needed)
//  2) the wave streams the selected 4KB opinion row -> out with 128-bit
//     non-temporal loads/stores (data is single-touch; opinions = 256MB > L2)
__global__ __launch_bounds__(32 * WAVES_PER_BLOCK)
void gather_argmax_kernel(const float* __restrict__ opinions,  // [E, B, D]
                          const float* __restrict__ weights,   // [B, E]
                          float* __restrict__ out)             // [B, D]
{
    const int lane = threadIdx.x & 31;
    const int wave = threadIdx.x >> 5;
    const int b    = blockIdx.x * WAVES_PER_BLOCK + wave;
    if (b >= B_) return;

    // ---- per-lane argmax over E=16 weights (4 x global_load_b128) ----
    const v4f* wrow = reinterpret_cast<const v4f*>(weights + (size_t)b * E_);
    v4f w0 = wrow[0];
    v4f w1 = wrow[1];
    v4f w2 = wrow[2];
    v4f w3 = wrow[3];
    float wv[E_] = { w0.x, w0.y, w0.z, w0.w,
                     w1.x, w1.y, w1.z, w1.w,
                     w2.x, w2.y, w2.z, w2.w,
                     w3.x, w3.y, w3.z, w3.w };
    int   best = 0;
    float bv   = wv[0];
#pragma unroll
    for (int i = 1; i < E_; ++i) {
        // strict '>' keeps the FIRST maximum, matching jnp.argmax tie-break
        if (wv[i] > bv) { bv = wv[i]; best = i; }
    }

    // ---- stream the selected row: 32 lanes x 8 v4f = 1024 floats ----
    const v4f* __restrict__ src =
        reinterpret_cast<const v4f*>(opinions + ((size_t)best * B_ + (size_t)b) * D_);
    v4f* __restrict__ dst =
        reinterpret_cast<v4f*>(out + (size_t)b * D_);

    // gfx1250 global_prefetch_b8: warm the data-dependent gather target
    __builtin_prefetch(src + lane, 0, 0);

    v4f v[F4_PER_LANE];
#pragma unroll
    for (int i = 0; i < F4_PER_LANE; ++i)
        v[i] = __builtin_nontemporal_load(src + lane + i * 32);
#pragma unroll
    for (int i = 0; i < F4_PER_LANE; ++i)
        __builtin_nontemporal_store(v[i], dst + lane + i * 32);
}

extern "C" void kernel_launch(void* const* d_in, const int* in_sizes, int n_in,
                              void* d_out, int out_size, void* d_ws, size_t ws_size,
                              hipStream_t stream) {
    const float* opinions = reinterpret_cast<const float*>(d_in[0]); // [E,B,D] f32
    const float* weights  = reinterpret_cast<const float*>(d_in[1]); // [B,E]   f32
    float*       out      = reinterpret_cast<float*>(d_out);         // [B,D]   f32

    const int blocks = B_ / WAVES_PER_BLOCK; // 512 blocks x 256 threads (8 waves)
    gather_argmax_kernel<<<blocks, 32 * WAVES_PER_BLOCK, 0, stream>>>(opinions, weights, out);
}